// SimplifiedGCN_17540646437559
// MI455X (gfx1250) — compile-verified
//
#include <hip/hip_runtime.h>
#include <hip/hip_bf16.h>
#include <math.h>

typedef float v2f __attribute__((ext_vector_type(2)));
typedef float v8f __attribute__((ext_vector_type(8)));

#define HIDDEN 128

// ---------------------------------------------------------------------------
// C[M,128] = relu(A[M,K] @ W[K,128] + b)   via V_WMMA_F32_16X16X4_F32
// 8 waves / block, each wave computes a 16-row strip x all 128 cols.
//
// A fragment (16x4 f32): lanes 0-15 hold K=k,k+1 ; lanes 16-31 hold K=k+2,k+3,
// row = lane&15  ->  contiguous 8-byte global load per lane.
//
// B fragments are staged in LDS pre-swizzled so each lane's (2-float) fragment
// is contiguous:  sW[((kg*8 + t)*32 + lane)*2 + e]
//                   = W[4*kg + (lane<16?0:2) + e][16*t + (lane&15)]
// -> one aligned ds_load_b64 per fragment, no register-packing movs.
//
// C tile: VGPR i -> row = tileBase + i (+8 for lanes 16-31), col = lane&15.
// ---------------------------------------------------------------------------
template <int KMAX>
__global__ __launch_bounds__(256) void gemm_relu_wmma(
    const float* __restrict__ A, const float* __restrict__ W,
    const float* __restrict__ bias, float* __restrict__ C,
    int M, int K) {
  constexpr int N = HIDDEN;
  __shared__ __align__(16) float sW[KMAX * N];  // swizzled fragment layout
  __shared__ __align__(16) float sB[N];

  // Cooperative load of W with swizzle into fragment layout, bias into LDS.
  {
    for (int i = threadIdx.x * 4; i < K * N; i += blockDim.x * 4) {
      const float4 w = *reinterpret_cast<const float4*>(W + i);
      const int kk = i >> 7;       // K row   (N == 128)
      const int n = i & 127;       // N col of w.x
      const int kg = kk >> 2;      // k-group (4 K values per WMMA)
      const int kr = kk & 3;       // position within k-group
      const int e = kr & 1;        // element within the 2-float fragment
      const int laneBase = (kr >> 1) ? 16 : 0;
      const float ws[4] = {w.x, w.y, w.z, w.w};
#pragma unroll
      for (int j = 0; j < 4; ++j) {
        const int nn = n + j;                     // stays inside the row
        const int t = nn >> 4;                    // n-tile
        const int nl = nn & 15;
        sW[((((kg << 3) + t) << 5) + laneBase + nl) * 2 + e] = ws[j];
      }
    }
    if (threadIdx.x < N) sB[threadIdx.x] = bias[threadIdx.x];
  }
  __syncthreads();

  const int lane = threadIdx.x & 31;
  const int wave = threadIdx.x >> 5;
  const int tile = blockIdx.x * 8 + wave;  // 16-row tile index
  const int rowTiles = M >> 4;             // M is a multiple of 16
  if (tile >= rowTiles) return;            // wave-uniform exit: EXEC all-1s

  const int nl = lane & 15;
  const int khalf = (lane < 16) ? 0 : 2;
  const int row = (tile << 4) + nl;
  const float* __restrict__ Arow = A + (size_t)row * (size_t)K;
  const v2f* __restrict__ sWf = reinterpret_cast<const v2f*>(sW);

  v8f acc[8];
#pragma unroll
  for (int t = 0; t < 8; ++t)
#pragma unroll
    for (int i = 0; i < 8; ++i) acc[t][i] = 0.0f;

  const int kGroups = K >> 2;
  for (int kg = 0; kg < kGroups; ++kg) {
    const v2f a = *reinterpret_cast<const v2f*>(Arow + (kg << 2) + khalf);
#pragma unroll
    for (int t = 0; t < 8; ++t) {
      const v2f b = sWf[(((kg << 3) + t) << 5) + lane];
      acc[t] = __builtin_amdgcn_wmma_f32_16x16x4_f32(
          false, a, false, b, (short)0, acc[t], false, false);
    }
  }

  // Bias + ReLU + store.
  const int rbase = (tile << 4) + ((lane < 16) ? 0 : 8);
#pragma unroll
  for (int t = 0; t < 8; ++t) {
    const int col = (t << 4) + nl;
    const float bv = sB[col];
#pragma unroll
    for (int i = 0; i < 8; ++i) {
      const float v = acc[t][i] + bv;
      C[(size_t)(rbase + i) * N + col] = fmaxf(v, 0.0f);
    }
  }
}

// ---------------------------------------------------------------------------
// agg[dst[e]][:] += h[src[e]][:]   (one wave per edge, 4 f32 atomics per lane)
// ---------------------------------------------------------------------------
__global__ __launch_bounds__(256) void scatter_add_kernel(
    const float* __restrict__ h, const int* __restrict__ src,
    const int* __restrict__ dst, float* __restrict__ agg, int nEdges) {
  const int e = blockIdx.x * 8 + (threadIdx.x >> 5);
  if (e >= nEdges) return;
  const int lane = threadIdx.x & 31;
  const int s = src[e];
  const int d = dst[e];
  const float* __restrict__ hs = h + (size_t)s * HIDDEN;
  float* __restrict__ ad = agg + (size_t)d * HIDDEN;
#pragma unroll
  for (int j = 0; j < 4; ++j) {
    const int c = lane + 32 * j;
    atomicAdd(&ad[c], hs[c]);
  }
}

// ---------------------------------------------------------------------------
// float4 grid-stride copy (initializes agg = h, folding the +h residual)
// ---------------------------------------------------------------------------
__global__ void copy_f4_kernel(const float* __restrict__ in,
                               float* __restrict__ out, size_t n4) {
  size_t i = (size_t)blockIdx.x * blockDim.x + threadIdx.x;
  const size_t stride = (size_t)gridDim.x * blockDim.x;
  const float4* __restrict__ s = reinterpret_cast<const float4*>(in);
  float4* __restrict__ d = reinterpret_cast<float4*>(out);
  for (; i < n4; i += stride) d[i] = s[i];
}

// ---------------------------------------------------------------------------
// out[p] = sigmoid( h[post[p]] . W_out + b_out )   (one wave per p)
// ---------------------------------------------------------------------------
__global__ __launch_bounds__(256) void readout_kernel(
    const float* __restrict__ h, const int* __restrict__ post,
    const float* __restrict__ Wout, const float* __restrict__ bout,
    float* __restrict__ out, int P) {
  const int p = blockIdx.x * 8 + (threadIdx.x >> 5);
  if (p >= P) return;
  const int lane = threadIdx.x & 31;
  const float* __restrict__ hr = h + (size_t)post[p] * HIDDEN;
  float sum = 0.0f;
#pragma unroll
  for (int j = 0; j < 4; ++j) {
    const int c = lane + 32 * j;
    sum += hr[c] * Wout[c];
  }
#pragma unroll
  for (int off = 16; off > 0; off >>= 1) sum += __shfl_down(sum, off, 32);
  if (lane == 0) {
    const float logit = sum + bout[0];
    out[p] = 1.0f / (1.0f + __expf(-logit));
  }
}

// ---------------------------------------------------------------------------
extern "C" void kernel_launch(void* const* d_in, const int* in_sizes, int n_in,
                              void* d_out, int out_size, void* d_ws,
                              size_t ws_size, hipStream_t stream) {
  const float* X      = (const float*)d_in[0];   // [M,256]
  const int*   eidx   = (const int*)d_in[1];     // [2,E]
  const int*   post   = (const int*)d_in[2];     // [P]
  const float* W_enc  = (const float*)d_in[3];   // [256,128]
  const float* b_enc  = (const float*)d_in[4];
  const float* W1     = (const float*)d_in[5];   // [128,128]
  const float* b1     = (const float*)d_in[6];
  const float* W2     = (const float*)d_in[7];
  const float* b2     = (const float*)d_in[8];
  const float* W_out  = (const float*)d_in[9];   // [128,1]
  const float* b_out  = (const float*)d_in[10];
  float* out = (float*)d_out;

  const int M = in_sizes[0] / 256;   // 100000
  const int E = in_sizes[1] / 2;     // 3200000
  const int P = in_sizes[2];         // 10000

  const size_t NH = (size_t)M * HIDDEN;
  float* h    = (float*)d_ws;        // [M,128]
  float* agg  = h + NH;              // [M,128]
  float* hnx  = agg + NH;            // [M,128]

  const int rowTiles = M >> 4;
  const int gemmBlocks = (rowTiles + 7) / 8;
  const int scatBlocks = (E + 7) / 8;
  const size_t n4 = NH >> 2;
  const int copyBlocks = 4096;

  const int* src = eidx;
  const int* dst = eidx + E;

  // h = relu(X @ W_enc + b_enc)            (K = 256)
  gemm_relu_wmma<256><<<gemmBlocks, 256, 0, stream>>>(X, W_enc, b_enc, h, M, 256);

  // agg = h ; agg += scatter(h) ; hnx = relu(agg @ W1 + b1)
  copy_f4_kernel<<<copyBlocks, 256, 0, stream>>>(h, agg, n4);
  scatter_add_kernel<<<scatBlocks, 256, 0, stream>>>(h, src, dst, agg, E);
  gemm_relu_wmma<128><<<gemmBlocks, 256, 0, stream>>>(agg, W1, b1, hnx, M, 128);

  // agg = hnx ; agg += scatter(hnx) ; h = relu(agg @ W2 + b2)
  copy_f4_kernel<<<copyBlocks, 256, 0, stream>>>(hnx, agg, n4);
  scatter_add_kernel<<<scatBlocks, 256, 0, stream>>>(hnx, src, dst, agg, E);
  gemm_relu_wmma<128><<<gemmBlocks, 256, 0, stream>>>(agg, W2, b2, h, M, 128);

  // out = sigmoid(h[post] @ W_out + b_out)
  readout_kernel<<<(P + 7) / 8, 256, 0, stream>>>(h, post, W_out, b_out, out, P);
}